// RecurrentAttentionCell_39539468927569
// MI455X (gfx1250) — compile-verified
//
#include <hip/hip_runtime.h>

typedef float v2f __attribute__((ext_vector_type(2)));
typedef float v8f __attribute__((ext_vector_type(8)));

#define NB 2
#define NH 16
#define NS 2048
#define ND 128
#define CK 64
#define NCHUNK (NS / CK)

// Skewed LDS layouts: conflict-free for both row reads and A-fragment (per-lane row) reads.
// 16B groups stay contiguous & aligned, so async B128 loads can target them directly.
#define SKK(r, c) (((r) << 7) + (((c) + 4 * (r)) & 127))
#define SKL(r, c) (((r) << 6) + (((c) + 4 * (r)) & 63))

// CDNA5 async global->LDS copy (ASYNCcnt), GVS mode: 64-bit SGPR base + 32-bit lane offset.
#define ASYNC_LDS_B128(ldsaddr, voff, sbase)                                   \
  asm volatile("global_load_async_to_lds_b128 %0, %1, %2"                      \
               ::"v"(ldsaddr), "v"(voff), "s"(sbase) : "memory")
#define WAIT_ASYNC() asm volatile("s_wait_asynccnt 0x0" ::: "memory")

__device__ __forceinline__ v8f wmma4(v2f a, v2f b, v8f c) {
  // D = A(16x4,f32) * B(4x16,f32) + C(16x16,f32)
  return __builtin_amdgcn_wmma_f32_16x16x4_f32(false, a, false, b, (short)0, c, false, false);
}

__global__ __launch_bounds__(256, 1)
void gdn_chunked_kernel(const float* __restrict__ qg, const float* __restrict__ kg,
                        const float* __restrict__ vg, const float* __restrict__ gg,
                        const float* __restrict__ bg, const float* __restrict__ s0g,
                        float* __restrict__ outg) {
  __shared__ float sS[128 * 128];  // recurrent state [k][v], row-major (row reads only)
  __shared__ float sK[CK * 128];   // K chunk, skewed
  __shared__ float sX[CK * 128];   // RHS -> D, solved in place (row reads only)
  __shared__ float sL[CK * CK];    // L matrix / its powers / later QK, skewed
  __shared__ float sc[CK];         // exp(+cumsum g)
  __shared__ float sci[CK];        // exp(-cumsum g)
  __shared__ float sb[CK];         // beta

  const int tid = threadIdx.x;
  const int lane = tid & 31;
  const int wid = tid >> 5;
  const int ln = lane & 15;          // fragment m/n index
  const int kb = (lane >> 4) << 1;   // fragment k base (0 or 2)
  const int rh = (lane >> 4) << 3;   // +8 row offset for C/D layout upper half

  const int bh = blockIdx.x;  // 0..31
  const long baseSD = (long)bh * NS * ND;
  const long baseS = (long)bh * NS;
  const long baseDD = (long)bh * ND * ND;

  // Initial state: async global->LDS, 16B per transfer.
  {
    const unsigned long long sbase = (unsigned long long)(s0g + baseDD);
    for (int i = tid; i < (128 * 128) / 4; i += 256) {
      const unsigned lds = (unsigned)(unsigned long long)(void*)&sS[i * 4];
      const unsigned goff = (unsigned)(i * 16);
      ASYNC_LDS_B128(lds, goff, sbase);
    }
    WAIT_ASYNC();
  }
  __syncthreads();

  for (int ci = 0; ci < NCHUNK; ++ci) {
    const int t0 = ci * CK;

    // Prefetch next chunk (one 256B line per thread per tensor) into cache.
    if (ci + 1 < NCHUNK && tid < 128) {
      const long nb = baseSD + (long)(t0 + CK) * ND + (long)tid * 64;
      __builtin_prefetch(kg + nb, 0, 3);
      __builtin_prefetch(qg + nb, 0, 3);
      __builtin_prefetch(vg + nb, 0, 3);
    }

    // Stage K chunk into skewed LDS via async copies (16B groups stay contiguous under skew).
    {
      const unsigned long long kbase = (unsigned long long)(kg + baseSD);
      for (int i = tid; i < (CK * ND) / 4; i += 256) {
        const int r = i >> 5;          // 32 x 16B per row
        const int c = (i & 31) * 4;
        const unsigned lds = (unsigned)(unsigned long long)(void*)&sK[SKK(r, c)];
        const unsigned goff = (unsigned)(((t0 + r) * ND + c) * 4);
        ASYNC_LDS_B128(lds, goff, kbase);
      }
    }
    // Gate prefix scan (inclusive cumsum of g) in wave 0 via shuffles.
    if (wid == 0) {
      const float g0 = gg[baseS + t0 + 2 * lane];
      const float g1 = gg[baseS + t0 + 2 * lane + 1];
      float s = g0 + g1;
#pragma unroll
      for (int d = 1; d < 32; d <<= 1) {
        const float t = __shfl_up(s, d, 32);
        if (lane >= d) s += t;
      }
      const float Go = s;       // G[2*lane+1]
      const float Ge = s - g1;  // G[2*lane]
      sc[2 * lane] = __expf(Ge);
      sc[2 * lane + 1] = __expf(Go);
      sci[2 * lane] = __expf(-Ge);
      sci[2 * lane + 1] = __expf(-Go);
    }
    if (tid < CK) sb[tid] = bg[baseS + t0 + tid];
    WAIT_ASYNC();
    __syncthreads();

    // ---- Phase A: X = beta .* (V - (c.*K) @ S0)   [64 x 128] ----
    {
      const int m0 = (wid >> 1) * 16;
      const int nb4 = (wid & 1) * 4;
      v8f acc[4];
      const v8f vz = {0.f, 0.f, 0.f, 0.f, 0.f, 0.f, 0.f, 0.f};
#pragma unroll
      for (int i = 0; i < 4; ++i) acc[i] = vz;
      const float ascl = sc[m0 + ln];
      for (int kk = 0; kk < ND; kk += 4) {
        v2f a;
        a.x = sK[SKK(m0 + ln, kk + kb)] * ascl;
        a.y = sK[SKK(m0 + ln, kk + kb + 1)] * ascl;
#pragma unroll
        for (int i = 0; i < 4; ++i) {
          const int n0 = (nb4 + i) * 16;
          v2f b;
          b.x = sS[(kk + kb) * 128 + n0 + ln];
          b.y = sS[(kk + kb + 1) * 128 + n0 + ln];
          acc[i] = wmma4(a, b, acc[i]);
        }
      }
#pragma unroll
      for (int i = 0; i < 4; ++i) {
        const int n0 = (nb4 + i) * 16;
#pragma unroll
        for (int r = 0; r < 8; ++r) {
          const int mr = m0 + r + rh;
          const float vv = vg[baseSD + (long)(t0 + mr) * ND + n0 + ln];
          sX[mr * 128 + n0 + ln] = sb[mr] * (vv - acc[i][r]);
        }
      }
    }

    // ---- Phase B: L = strict_tril( (beta.*c.*K) @ (K./c)^T )   [64 x 64] ----
    {
      const int m0 = (wid >> 1) * 16;
      const int nb2 = (wid & 1) * 2;
      v8f acc[2];
      const v8f vz = {0.f, 0.f, 0.f, 0.f, 0.f, 0.f, 0.f, 0.f};
      acc[0] = vz; acc[1] = vz;
      const float ascl = sc[m0 + ln] * sb[m0 + ln];
      float bscl[2];
#pragma unroll
      for (int i = 0; i < 2; ++i) bscl[i] = sci[(nb2 + i) * 16 + ln];
      for (int kk = 0; kk < ND; kk += 4) {
        v2f a;
        a.x = sK[SKK(m0 + ln, kk + kb)] * ascl;
        a.y = sK[SKK(m0 + ln, kk + kb + 1)] * ascl;
#pragma unroll
        for (int i = 0; i < 2; ++i) {
          const int n0 = (nb2 + i) * 16;
          v2f b;
          b.x = sK[SKK(n0 + ln, kk + kb)] * bscl[i];
          b.y = sK[SKK(n0 + ln, kk + kb + 1)] * bscl[i];
          acc[i] = wmma4(a, b, acc[i]);
        }
      }
#pragma unroll
      for (int i = 0; i < 2; ++i) {
        const int n0 = (nb2 + i) * 16;
#pragma unroll
        for (int r = 0; r < 8; ++r) {
          const int mr = m0 + r + rh;
          sL[SKL(mr, n0 + ln)] = (mr > n0 + ln) ? acc[i][r] : 0.0f;
        }
      }
    }
    __syncthreads();

    // ---- Phase C: X <- (I+L)^-1 X via commuting nilpotent factors ----
    auto applyL = [&](float sgn) {  // X <- X + sgn * L @ X (in place, two-phase)
      const int m0 = (wid >> 1) * 16;
      const int nb4 = (wid & 1) * 4;
      v8f acc[4];
#pragma unroll
      for (int i = 0; i < 4; ++i) {
        const int n0 = (nb4 + i) * 16;
#pragma unroll
        for (int r = 0; r < 8; ++r) acc[i][r] = sX[(m0 + r + rh) * 128 + n0 + ln];
      }
      for (int kk = 0; kk < CK; kk += 4) {
        v2f a;
        a.x = sL[SKL(m0 + ln, kk + kb)] * sgn;
        a.y = sL[SKL(m0 + ln, kk + kb + 1)] * sgn;
#pragma unroll
        for (int i = 0; i < 4; ++i) {
          const int n0 = (nb4 + i) * 16;
          v2f b;
          b.x = sX[(kk + kb) * 128 + n0 + ln];
          b.y = sX[(kk + kb + 1) * 128 + n0 + ln];
          acc[i] = wmma4(a, b, acc[i]);
        }
      }
      __syncthreads();
#pragma unroll
      for (int i = 0; i < 4; ++i) {
        const int n0 = (nb4 + i) * 16;
#pragma unroll
        for (int r = 0; r < 8; ++r) sX[(m0 + r + rh) * 128 + n0 + ln] = acc[i][r];
      }
      __syncthreads();
    };
    auto squareL = [&]() {  // L <- L @ L (in place, two-phase)
      const int m0 = (wid >> 1) * 16;
      const int nb2 = (wid & 1) * 2;
      v8f acc[2];
      const v8f vz = {0.f, 0.f, 0.f, 0.f, 0.f, 0.f, 0.f, 0.f};
      acc[0] = vz; acc[1] = vz;
      for (int kk = 0; kk < CK; kk += 4) {
        v2f a;
        a.x = sL[SKL(m0 + ln, kk + kb)];
        a.y = sL[SKL(m0 + ln, kk + kb + 1)];
#pragma unroll
        for (int i = 0; i < 2; ++i) {
          const int n0 = (nb2 + i) * 16;
          v2f b;
          b.x = sL[SKL(kk + kb, n0 + ln)];
          b.y = sL[SKL(kk + kb + 1, n0 + ln)];
          acc[i] = wmma4(a, b, acc[i]);
        }
      }
      __syncthreads();
#pragma unroll
      for (int i = 0; i < 2; ++i) {
        const int n0 = (nb2 + i) * 16;
#pragma unroll
        for (int r = 0; r < 8; ++r) sL[SKL(m0 + r + rh, n0 + ln)] = acc[i][r];
      }
      __syncthreads();
    };

    applyL(-1.0f);                    // (I - L)
    for (int it = 0; it < 5; ++it) {  // (I + L^2)(I + L^4)(I + L^8)(I + L^16)(I + L^32)
      squareL();
      applyL(1.0f);
    }
    // sX now holds D.

    // ---- Phase D1: QK = incl_tril( (c.*Q) @ (K./c)^T )  (reuses sL) ----
    {
      const int m0 = (wid >> 1) * 16;
      const int nb2 = (wid & 1) * 2;
      v8f acc[2];
      const v8f vz = {0.f, 0.f, 0.f, 0.f, 0.f, 0.f, 0.f, 0.f};
      acc[0] = vz; acc[1] = vz;
      const float ascl = sc[m0 + ln];
      const long qrow = baseSD + (long)(t0 + m0 + ln) * ND;
      float bscl[2];
#pragma unroll
      for (int i = 0; i < 2; ++i) bscl[i] = sci[(nb2 + i) * 16 + ln];
      for (int kk = 0; kk < ND; kk += 4) {
        v2f a;
        a.x = qg[qrow + kk + kb] * ascl;
        a.y = qg[qrow + kk + kb + 1] * ascl;
#pragma unroll
        for (int i = 0; i < 2; ++i) {
          const int n0 = (nb2 + i) * 16;
          v2f b;
          b.x = sK[SKK(n0 + ln, kk + kb)] * bscl[i];
          b.y = sK[SKK(n0 + ln, kk + kb + 1)] * bscl[i];
          acc[i] = wmma4(a, b, acc[i]);
        }
      }
#pragma unroll
      for (int i = 0; i < 2; ++i) {
        const int n0 = (nb2 + i) * 16;
#pragma unroll
        for (int r = 0; r < 8; ++r) {
          const int mr = m0 + r + rh;
          sL[SKL(mr, n0 + ln)] = (mr >= n0 + ln) ? acc[i][r] : 0.0f;
        }
      }
    }
    __syncthreads();

    // ---- Phase D2: O = (c.*Q) @ S0 + QK @ D  -> global out ----
    {
      const int m0 = (wid >> 1) * 16;
      const int nb4 = (wid & 1) * 4;
      v8f acc[4];
      const v8f vz = {0.f, 0.f, 0.f, 0.f, 0.f, 0.f, 0.f, 0.f};
#pragma unroll
      for (int i = 0; i < 4; ++i) acc[i] = vz;
      const float ascl = sc[m0 + ln];
      const long qrow = baseSD + (long)(t0 + m0 + ln) * ND;
      for (int kk = 0; kk < ND; kk += 4) {
        v2f a;
        a.x = qg[qrow + kk + kb] * ascl;
        a.y = qg[qrow + kk + kb + 1] * ascl;
#pragma unroll
        for (int i = 0; i < 4; ++i) {
          const int n0 = (nb4 + i) * 16;
          v2f b;
          b.x = sS[(kk + kb) * 128 + n0 + ln];
          b.y = sS[(kk + kb + 1) * 128 + n0 + ln];
          acc[i] = wmma4(a, b, acc[i]);
        }
      }
      for (int kk = 0; kk < CK; kk += 4) {
        v2f a;
        a.x = sL[SKL(m0 + ln, kk + kb)];
        a.y = sL[SKL(m0 + ln, kk + kb + 1)];
#pragma unroll
        for (int i = 0; i < 4; ++i) {
          const int n0 = (nb4 + i) * 16;
          v2f b;
          b.x = sX[(kk + kb) * 128 + n0 + ln];
          b.y = sX[(kk + kb + 1) * 128 + n0 + ln];
          acc[i] = wmma4(a, b, acc[i]);
        }
      }
#pragma unroll
      for (int i = 0; i < 4; ++i) {
        const int n0 = (nb4 + i) * 16;
#pragma unroll
        for (int r = 0; r < 8; ++r) {
          const int mr = m0 + r + rh;
          outg[baseSD + (long)(t0 + mr) * ND + n0 + ln] = acc[i][r];
        }
      }
    }

    // ---- Phase D3: S <- cC*S + (cC * K./c)^T @ D   [128 x 128] ----
    {
      const float cC = sc[CK - 1];
      const int m0 = wid * 16;
      v8f acc[8];
#pragma unroll
      for (int i = 0; i < 8; ++i) {
        const int n0 = i * 16;
#pragma unroll
        for (int r = 0; r < 8; ++r) acc[i][r] = cC * sS[(m0 + r + rh) * 128 + n0 + ln];
      }
      for (int kk = 0; kk < CK; kk += 4) {
        v2f a;
        a.x = sK[SKK(kk + kb, m0 + ln)] * (cC * sci[kk + kb]);
        a.y = sK[SKK(kk + kb + 1, m0 + ln)] * (cC * sci[kk + kb + 1]);
#pragma unroll
        for (int i = 0; i < 8; ++i) {
          const int n0 = i * 16;
          v2f b;
          b.x = sX[(kk + kb) * 128 + n0 + ln];
          b.y = sX[(kk + kb + 1) * 128 + n0 + ln];
          acc[i] = wmma4(a, b, acc[i]);
        }
      }
      __syncthreads();  // everyone done reading old S before overwrite
#pragma unroll
      for (int i = 0; i < 8; ++i) {
        const int n0 = i * 16;
#pragma unroll
        for (int r = 0; r < 8; ++r) sS[(m0 + r + rh) * 128 + n0 + ln] = acc[i][r];
      }
      __syncthreads();
    }
  }

  // Final recurrent state, appended after the attention output.
  const long stateOut = (long)NB * NH * NS * ND + baseDD;
  for (int i = tid; i < 128 * 128; i += 256) outg[stateOut + i] = sS[i];
}

extern "C" void kernel_launch(void* const* d_in, const int* in_sizes, int n_in,
                              void* d_out, int out_size, void* d_ws, size_t ws_size,
                              hipStream_t stream) {
  (void)in_sizes; (void)n_in; (void)out_size; (void)d_ws; (void)ws_size;
  const float* q = (const float*)d_in[0];
  const float* k = (const float*)d_in[1];
  const float* v = (const float*)d_in[2];
  const float* g = (const float*)d_in[3];
  const float* b = (const float*)d_in[4];
  const float* s0 = (const float*)d_in[5];
  float* out = (float*)d_out;
  gdn_chunked_kernel<<<dim3(NB * NH), dim3(256), 0, stream>>>(q, k, v, g, b, s0, out);
}